// Net_14396730376572
// MI455X (gfx1250) — compile-verified
//
#include <hip/hip_runtime.h>
#include <math.h>

#define NN 12000
#define EE 400000

typedef __attribute__((ext_vector_type(2))) float v2f;
typedef __attribute__((ext_vector_type(8))) float v8f;

// ---------------------------------------------------------------- utilities
__global__ void fill_kernel(float* __restrict__ p, float v, int n) {
    int i = blockIdx.x * blockDim.x + threadIdx.x;
    if (i < n) p[i] = v;
}

// Degrees for both normalizations in one pass.
// dense path: out-degree of src over non-self edges (+1 diag preset)
// sparse path: in-degree of dst over ALL edges (+1 self-loop preset)
__global__ void deg_kernel(const int* __restrict__ src, const int* __restrict__ dst,
                           float* degD, float* degS, int E) {
    int e = blockIdx.x * blockDim.x + threadIdx.x;
    if (e >= E) return;
    int s = src[e], d = dst[e];
    if (s != d) atomicAdd(&degD[s], 1.0f);
    atomicAdd(&degS[d], 1.0f);
}

__global__ void rsqrt_kernel(float* __restrict__ p, int n) {
    int i = blockIdx.x * blockDim.x + threadIdx.x;
    if (i < n) p[i] = rsqrtf(p[i]);
}

// Pad W2 [K x dout] into [K x Npad] with zero columns so the GEMM N-dim is a
// multiple of 16 (no divergent guards around WMMA).
__global__ void padw_kernel(const float* __restrict__ W, float* __restrict__ Wp,
                            int K, int dout, int Npad) {
    int t = blockIdx.x * blockDim.x + threadIdx.x;
    if (t >= K * Npad) return;
    int k = t / Npad, c = t - k * Npad;
    Wp[t] = (c < dout) ? W[k * dout + c] : 0.0f;
}

// ------------------------------------------------- WMMA f32 GEMM + row scale
// Hs[m, n] = rowScale[m] * sum_k X[m,k] * W[k,n]
// One wave per 16x16 tile, V_WMMA_F32_16X16X4_F32 stepping K by 4.
// Requires: M % 16 == 0, K % 4 == 0, Npad % 16 == 0 (all hold here).
__global__ void __launch_bounds__(256)
gemm_scale_wmma(const float* __restrict__ X, const float* __restrict__ W,
                float* __restrict__ Hs, const float* __restrict__ rowScale,
                int M, int K, int Npad) {
    int wave = blockIdx.x * (blockDim.x >> 5) + (threadIdx.x >> 5);
    int lane = threadIdx.x & 31;                 // wave32
    int ntn  = Npad >> 4;
    int totalTiles = (M >> 4) * ntn;
    if (wave >= totalTiles) return;              // wave-uniform: EXEC stays all-1s
    int tm = wave / ntn;
    int tn = wave - tm * ntn;
    int half = lane >> 4;                        // 0: K=k0,k0+1  1: K=k0+2,k0+3
    int l    = lane & 15;
    int arow = (tm << 4) + l;                    // A row for this lane
    int bcol = (tn << 4) + l;                    // B/D column for this lane
    const float* Xr = X + (size_t)arow * K;

    v8f c = {};
#pragma unroll 4
    for (int k0 = 0; k0 < K; k0 += 4) {
        int ka = k0 + (half << 1);
        v2f a;                                   // A 16x4 f32 fragment
        a.x = Xr[ka];
        a.y = Xr[ka + 1];
        v2f b;                                   // B 4x16 f32 fragment
        b.x = W[(size_t)ka * Npad + bcol];
        b.y = W[(size_t)(ka + 1) * Npad + bcol];
        // (neg_a, A, neg_b, B, c_mod, C, reuse_a, reuse_b)
        c = __builtin_amdgcn_wmma_f32_16x16x4_f32(false, a, false, b,
                                                  (short)0, c, false, false);
    }

    int rbase = (tm << 4) + (half << 3);         // C/D: VGPR v -> row v (+8 for hi half)
#pragma unroll
    for (int v = 0; v < 8; ++v) {
        int row = rbase + v;
        Hs[(size_t)row * Npad + bcol] = c[v] * rowScale[row];
    }
}

// ------------------------------------------------------ edge scatter-add
// denseMode=1: Acc[src] += Hs[dst]  (skip self edges; diag handled separately)
// denseMode=0: Acc[dst] += Hs[src]  (all edges)
// One thread per (edge, channel); dout is a power of two (shift sh).
__global__ void edge_agg_kernel(const int* __restrict__ src, const int* __restrict__ dst,
                                const float* __restrict__ Hs, float* __restrict__ Acc,
                                int E, int sh, int denseMode) {
    long long total  = (long long)E << sh;
    int       mask   = (1 << sh) - 1;
    long long stride = (long long)gridDim.x * blockDim.x;
    for (long long t = (long long)blockIdx.x * blockDim.x + threadIdx.x;
         t < total; t += stride) {
        int e = (int)(t >> sh);
        int c = (int)t & mask;
        int s = src[e], d = dst[e];
        if (denseMode) {
            if (s != d)
                atomicAdd(&Acc[((size_t)s << sh) + c], Hs[((size_t)d << sh) + c]);
        } else {
            atomicAdd(&Acc[((size_t)d << sh) + c], Hs[((size_t)s << sh) + c]);
        }
    }
}

// out[i,c] = rowScale[i]*(Acc[i,c] + Hs[i,c]) + bias[c]   (+ReLU)
__global__ void finalize_kernel(const float* __restrict__ Acc, const float* __restrict__ Hs,
                                const float* __restrict__ rowScale, const float* __restrict__ bias,
                                float* __restrict__ Out, int Nn, int sh, int doRelu) {
    int t = blockIdx.x * blockDim.x + threadIdx.x;
    int total = Nn << sh;
    if (t >= total) return;
    int i = t >> sh;
    int c = t & ((1 << sh) - 1);
    float v = rowScale[i] * (Acc[t] + Hs[t]) + bias[c];
    if (doRelu) v = fmaxf(v, 0.0f);
    Out[t] = v;
}

// Fused last-layer epilogue + log_softmax over C=6 (Hs/Acc strided by 16).
__global__ void final_logsoftmax(const float* __restrict__ Acc, const float* __restrict__ Hs,
                                 const float* __restrict__ rowScale, const float* __restrict__ bias,
                                 float* __restrict__ Out, int Nn) {
    int i = blockIdx.x * blockDim.x + threadIdx.x;
    if (i >= Nn) return;
    float sc = rowScale[i];
    float v[6];
    float mx = -INFINITY;
#pragma unroll
    for (int c = 0; c < 6; ++c) {
        float x = sc * (Acc[i * 16 + c] + Hs[i * 16 + c]) + bias[c];
        v[c] = x;
        mx = fmaxf(mx, x);
    }
    float s = 0.0f;
#pragma unroll
    for (int c = 0; c < 6; ++c) s += expf(v[c] - mx);
    float ls = logf(s);
#pragma unroll
    for (int c = 0; c < 6; ++c) Out[i * 6 + c] = v[c] - mx - ls;
}

// ---------------------------------------------------------------- driver
extern "C" void kernel_launch(void* const* d_in, const int* in_sizes, int n_in,
                              void* d_out, int out_size, void* d_ws, size_t ws_size,
                              hipStream_t stream) {
    const float* x0  = (const float*)d_in[0];
    const int*   ei  = (const int*)d_in[1];
    const int*   src = ei;
    const int*   dst = ei + EE;
    const float* W1  = (const float*)d_in[2];  const float* b1  = (const float*)d_in[3];
    const float* W12 = (const float*)d_in[4];  const float* b12 = (const float*)d_in[5];
    const float* W13 = (const float*)d_in[6];  const float* b13 = (const float*)d_in[7];
    const float* W14 = (const float*)d_in[8];  const float* b14 = (const float*)d_in[9];
    const float* W2  = (const float*)d_in[10]; const float* b2  = (const float*)d_in[11];
    float* out = (float*)d_out;

    // workspace layout (floats): ~24.7 MB total
    float* ws   = (float*)d_ws;
    float* dis  = ws;                         // [N]  dense-path  deg^-1/2
    float* dinv = ws + NN;                    // [N]  sparse-path deg^-1/2
    float* Hs   = ws + 2 * NN;                // [N x 128] scaled h
    float* Acc  = Hs  + (size_t)NN * 128;     // [N x 128] edge accumulator
    float* xa   = Acc + (size_t)NN * 128;     // [N x 128] activation ping
    float* xb   = xa  + (size_t)NN * 128;     // [N x 128] activation pong
    float* W2p  = xb  + (size_t)NN * 128;     // [128 x 16] zero-padded W2

    // --- degrees -> rsqrt scales (dis, dinv contiguous: one fill, one rsqrt)
    fill_kernel<<<(2 * NN + 255) / 256, 256, 0, stream>>>(dis, 1.0f, 2 * NN);
    deg_kernel<<<(EE + 255) / 256, 256, 0, stream>>>(src, dst, dis, dinv, EE);
    rsqrt_kernel<<<(2 * NN + 255) / 256, 256, 0, stream>>>(dis, 2 * NN);
    padw_kernel<<<(128 * 16 + 255) / 256, 256, 0, stream>>>(W2, W2p, 128, 6, 16);

    auto run_layer = [&](const float* X, int K, const float* Wm, const float* bias,
                         float* Xout, int dout, int sh, const float* scale,
                         int denseMode, int doRelu) {
        int tiles = (NN / 16) * (dout / 16);
        gemm_scale_wmma<<<(tiles + 7) / 8, 256, 0, stream>>>(X, Wm, Hs, scale, NN, K, dout);
        fill_kernel<<<((NN << sh) + 255) / 256, 256, 0, stream>>>(Acc, 0.0f, NN << sh);
        long long total = (long long)EE << sh;
        long long want  = (total + 255) / 256;
        int blocks = want > 8192 ? 8192 : (int)want;
        edge_agg_kernel<<<blocks, 256, 0, stream>>>(src, dst, Hs, Acc, EE, sh, denseMode);
        if (Xout)
            finalize_kernel<<<((NN << sh) + 255) / 256, 256, 0, stream>>>(
                Acc, Hs, scale, bias, Xout, NN, sh, doRelu);
    };

    run_layer(x0, 512, W1,  b1,  xa, 32,  5, dis,  1, 1);   // dense GCN 512->32
    run_layer(xa, 32,  W12, b12, xb, 64,  6, dis,  1, 1);   // dense GCN 32->64
    run_layer(xb, 64,  W13, b13, xa, 128, 7, dinv, 0, 1);   // sparse GCN 64->128
    run_layer(xa, 128, W14, b14, xb, 128, 7, dinv, 0, 1);   // sparse GCN 128->128
    run_layer(xb, 128, W2p, nullptr, nullptr, 16, 4, dinv, 0, 0); // sparse 128->6 (padded)

    final_logsoftmax<<<(NN + 255) / 256, 256, 0, stream>>>(Acc, Hs, dinv, b2, out, NN);
}